// NotSimpleAgg_57157424775594
// MI455X (gfx1250) — compile-verified
//
#include <hip/hip_runtime.h>

typedef float v2f __attribute__((ext_vector_type(2)));
typedef float v8f __attribute__((ext_vector_type(8)));

#define NROW 32
#define NC 1024
#define LSTRIDE 1028   // padded row stride (floats); 1028 % 64 = 4 -> conflict-free WMMA row gathers
#define NSLOT 33       // 32 rows + 1 free slot for the freshly merged row

__global__ __launch_bounds__(128) void agg_merge_kernel(const float* __restrict__ x,
                                                        const float* __restrict__ convw,
                                                        const float* __restrict__ convb,
                                                        float* __restrict__ out) {
  __shared__ __align__(16) float xs[NSLOT * LSTRIDE];   // row data, physical slots
  __shared__ float G[NSLOT * NSLOT];                    // Gram (dot products) over physical slots
  __shared__ int   lmap[NROW];                          // logical row -> physical slot
  __shared__ int   perm[NROW];
  __shared__ int   sA, sB, sFree;

  const int b    = blockIdx.x;
  const int tid  = threadIdx.x;
  const int wave = tid >> 5;
  const int lane = tid & 31;

  // conv weights: conv_w shape (1,2,3) row-major, conv_b scalar
  const float w00 = convw[0], w01 = convw[1], w02 = convw[2];
  const float w10 = convw[3], w11 = convw[4], w12 = convw[5];
  const float bias = convb[0];

  // ---- load this batch's 32x1024 fp32 rows into LDS (coalesced float4) ----
  const float4* xg = (const float4*)(x + (size_t)b * NROW * NC);
  for (int v = tid; v < NROW * NC / 4; v += 128) {
    const int e = v << 2;
    const int row = e >> 10;
    const int col = e & (NC - 1);
    float4 val = xg[v];
    float* p = &xs[row * LSTRIDE + col];
    p[0] = val.x; p[1] = val.y; p[2] = val.z; p[3] = val.w;
  }
  if (tid < NROW) lmap[tid] = tid;
  if (tid == 0) sFree = NROW;
  __syncthreads();

  // ---- initial 32x32 Gram via V_WMMA_F32_16X16X4_F32: 4 waves, one 16x16 tile each ----
  {
    const int ti = (wave >> 1) * 16;     // tile row base
    const int tj = (wave & 1) * 16;      // tile col base
    const int mr = ti + (lane & 15);     // A: M = lane&15
    const int nc = tj + (lane & 15);     // B: N = lane&15 (symmetric layout)
    const int ko = (lane >> 4) * 2;      // lanes 16-31 carry K=2,3
    v8f c = {0.f, 0.f, 0.f, 0.f, 0.f, 0.f, 0.f, 0.f};
    for (int k0 = 0; k0 < NC; k0 += 4) {
      v2f A = *(const v2f*)&xs[mr * LSTRIDE + k0 + ko];
      v2f Bf = *(const v2f*)&xs[nc * LSTRIDE + k0 + ko];
      c = __builtin_amdgcn_wmma_f32_16x16x4_f32(false, A, false, Bf, (short)0, c, false, false);
    }
    for (int r = 0; r < 8; ++r) {        // C/D layout: M = r + 8*(lane>>4), N = lane&15
      const int gm = ti + r + ((lane >> 4) << 3);
      const int gn = tj + (lane & 15);
      G[gm * NSLOT + gn] = c[r];
    }
  }
  __syncthreads();

  // ---- 31 sequential merge steps ----
  for (int ncur = NROW; ncur >= 2; --ncur) {
    // Phase A: argmin over logical pairs i<j (wave 0); tie-break = lexicographic (i,j)
    if (wave == 0) {
      unsigned long long best = ~0ull;
      for (int idx = lane; idx < ncur * ncur; idx += 32) {
        const int i = idx / ncur;
        const int j = idx - i * ncur;
        if (j <= i) continue;
        const int pi = lmap[i], pj = lmap[j];
        float d2 = G[pi * NSLOT + pi] + G[pj * NSLOT + pj] - 2.f * G[pi * NSLOT + pj];
        d2 = fmaxf(d2, 0.f);  // matches sqrt(max(d2,0)) ordering & ties
        const unsigned long long key =
            ((unsigned long long)__float_as_uint(d2) << 10) | (unsigned)((i << 5) | j);
        if (key < best) best = key;
      }
      for (int m = 16; m >= 1; m >>= 1) {
        unsigned long long o = __shfl_xor(best, m, 32);
        if (o < best) best = o;
      }
      if (lane == 0) { sA = (int)((best >> 5) & 31); sB = (int)(best & 31); }
    }
    __syncthreads();
    const int a  = sA;
    const int bb = sB;
    const int pa = lmap[a], pb = lmap[bb];
    const int F  = sFree;

    // Phase B: 3-tap SAME conv of rows (pa, pb) + bias + ReLU into free slot F
    for (int r = 0; r < 8; ++r) {
      const int h = tid + (r << 7);
      const float am1 = (h > 0)      ? xs[pa * LSTRIDE + h - 1] : 0.f;
      const float a0  =               xs[pa * LSTRIDE + h];
      const float ap1 = (h < NC - 1) ? xs[pa * LSTRIDE + h + 1] : 0.f;
      const float bm1 = (h > 0)      ? xs[pb * LSTRIDE + h - 1] : 0.f;
      const float b0  =               xs[pb * LSTRIDE + h];
      const float bp1 = (h < NC - 1) ? xs[pb * LSTRIDE + h + 1] : 0.f;
      const float m = bias + w00 * am1 + w01 * a0 + w02 * ap1
                           + w10 * bm1 + w11 * b0 + w12 * bp1;
      xs[F * LSTRIDE + h] = fmaxf(m, 0.f);
    }

    // Exact reference permutation semantics (including the a==1 drop-row-0 quirk)
    if (tid < ncur) perm[tid] = tid;
    __syncthreads();
    if (tid == 0) { perm[a] = 0; perm[0] = a; perm[bb] = 1; perm[1] = bb; }
    __syncthreads();
    int tmpv = 0;
    const bool doMap = (tid >= 1 && tid <= ncur - 2);
    if (doMap) tmpv = lmap[perm[tid + 1]];
    __syncthreads();
    if (doMap) lmap[tid] = tmpv;
    if (tid == 0) { lmap[0] = F; sFree = pb; }  // oldmap[bb] is free in every case
    __syncthreads();

    // Phase C: dots of merged row vs all new live rows (t==0 -> self dot = new diag entry)
    for (int t = wave; t <= ncur - 2; t += 4) {
      const int q = lmap[t];
      float acc = 0.f;
      for (int h = lane; h < NC; h += 32)
        acc += xs[F * LSTRIDE + h] * xs[q * LSTRIDE + h];
      for (int m = 16; m >= 1; m >>= 1)
        acc += __shfl_xor(acc, m, 32);
      if (lane == 0) { G[F * NSLOT + q] = acc; G[q * NSLOT + F] = acc; }
    }
    __syncthreads();
  }

  // ---- final row out ----
  const int fin = lmap[0];
  float* o = out + (size_t)b * NC;
  for (int h = tid; h < NC; h += 128) o[h] = xs[fin * LSTRIDE + h];
}

extern "C" void kernel_launch(void* const* d_in, const int* in_sizes, int n_in,
                              void* d_out, int out_size, void* d_ws, size_t ws_size,
                              hipStream_t stream) {
  const float* x  = (const float*)d_in[0];   // (512, 32, 1024) fp32
  const float* cw = (const float*)d_in[1];   // (1, 2, 3) fp32
  const float* cb = (const float*)d_in[2];   // (1,) fp32
  float* out = (float*)d_out;                // (512, 1024) fp32
  const int nb = in_sizes[0] / (NROW * NC);
  agg_merge_kernel<<<nb, 128, 0, stream>>>(x, cw, cb, out);
}